// BasicBlock_3058016715396
// MI455X (gfx1250) — compile-verified
//
#include <hip/hip_runtime.h>

typedef float v2f __attribute__((ext_vector_type(2)));
typedef float v8f __attribute__((ext_vector_type(8)));

#define WAVES_PER_BLOCK 8

// Repack W[k][c][d] (64x64 per k) into lane-major, operand-pair-contiguous tiles:
//   Wp[((k*16 + cs)*2 + hi)*128 + l15*8 + (dt*2 + kk)]
//     = W[k][cs*4 + 2*hi + kk][l15 + 16*dt]
// so one float4 load yields B0=(q.x,q.y), B1=(q.z,q.w) as adjacent VGPR pairs
// (no shuffle movs before the WMMA).
__global__ __launch_bounds__(256)
void repack_W(const float* __restrict__ W, float* __restrict__ Wp, int total)
{
    int t = blockIdx.x * blockDim.x + threadIdx.x;
    if (t >= total) return;
    const int j   = t & 7;
    const int kk  = j & 1;
    const int dt  = j >> 1;
    const int l15 = (t >> 3) & 15;
    const int hi  = (t >> 7) & 1;
    const int cs  = (t >> 8) & 15;
    const int k   = t >> 12;
    const int c   = cs * 4 + 2 * hi + kk;
    const int d   = l15 + 16 * dt;
    Wp[t] = W[((size_t)k * 64 + c) * 64 + d];
}

// D = relu( sum_k mask(n,k) * W[k]^T @ feats[idx[n,k]]  (+ resid) )
// One wave32 computes TWO 16x64 output tiles with V_WMMA_F32_16X16X4_F32.
// Masking is done by redirecting the gather to a zero row (no per-element VALU).
template <bool FUSE_RESIDUAL>
__global__ __launch_bounds__(256)
void sparse_conv_wmma(const float* __restrict__ feats,  // [N,64] features to gather
                      const int*   __restrict__ idx,    // [N,K]
                      const int*   __restrict__ mask,   // [N,K] (0/1)
                      const float* __restrict__ Wp,     // repacked weights [K*4096]
                      const float* __restrict__ zrow,   // 64 zeros
                      const float* __restrict__ resid,  // [N,64] or nullptr
                      float*       __restrict__ out,    // [N,64]
                      int N, int K)
{
    const int lane = threadIdx.x & 31;
    const int wave = threadIdx.x >> 5;
    const int pair = blockIdx.x * WAVES_PER_BLOCK + wave;
    const int numTiles = N >> 4;                 // 100000/16 = 6250
    const int tile0 = pair * 2;
    if (tile0 >= numTiles) return;               // wave-uniform: EXEC stays all-1s
    const bool has1 = (tile0 + 1) < numTiles;    // wave-uniform

    const int l15 = lane & 15;                   // matrix column / row-in-tile
    const int hi  = lane >> 4;                   // half-wave selects K-pair {0,1} vs {2,3}
    const int row0 = tile0 * 16 + l15;
    const int row1 = has1 ? row0 + 16 : row0;    // degenerate (harmless) when no 2nd tile

    v8f a0a = {}, a1a = {}, a2a = {}, a3a = {};  // tile0 accumulators (4 d-tiles)
    v8f a0b = {}, a1b = {}, a2b = {}, a3b = {};  // tile1 accumulators

    // per-lane base into repacked weights: + k*4096 + cs*256 each step
    const float* wk = Wp + (size_t)hi * 128 + (size_t)l15 * 8;

    // software-pipelined idx/mask (one load per (row,k), also feeds prefetch)
    int s0 = idx[(size_t)row0 * K];
    int s1 = idx[(size_t)row1 * K];
    int m0 = mask[(size_t)row0 * K];
    int m1 = mask[(size_t)row1 * K];

    for (int k = 0; k < K; ++k) {
        // mask==0  ==>  gather the zero row instead of multiplying by 0
        const float* __restrict__ g0 = m0 ? feats + (size_t)s0 * 64 : zrow;
        const float* __restrict__ g1 = m1 ? feats + (size_t)s1 * 64 : zrow;
        const float* __restrict__ w  = wk + (size_t)k * 4096;

        if (k + 1 < K) {
            s0 = idx[(size_t)row0 * K + k + 1];
            s1 = idx[(size_t)row1 * K + k + 1];
            m0 = mask[(size_t)row0 * K + k + 1];
            m1 = mask[(size_t)row1 * K + k + 1];
            __builtin_prefetch(feats + (size_t)s0 * 64, 0, 3);  // warm L0 for next gather
            __builtin_prefetch(feats + (size_t)s1 * 64, 0, 3);
        }

#pragma unroll
        for (int cs = 0; cs < 16; ++cs) {
            // A operands: 16x4 f32; lane<16 K={c0,c0+1}, lane>=16 K={c0+2,c0+3}
            const v2f A0 = *(const v2f*)(g0 + cs * 4 + 2 * hi);
            const v2f A1 = *(const v2f*)(g1 + cs * 4 + 2 * hi);

            // B operands: 2x coalesced b128; pairs land in adjacent VGPRs
            const float4 q0 = *(const float4*)(w + (size_t)cs * 256);      // B0,B1
            const float4 q1 = *(const float4*)(w + (size_t)cs * 256 + 4);  // B2,B3
            v2f B0; B0.x = q0.x; B0.y = q0.y;
            v2f B1; B1.x = q0.z; B1.y = q0.w;
            v2f B2; B2.x = q1.x; B2.y = q1.y;
            v2f B3; B3.x = q1.z; B3.y = q1.w;

            a0a = __builtin_amdgcn_wmma_f32_16x16x4_f32(false, A0, false, B0, (short)0, a0a, false, false);
            a0b = __builtin_amdgcn_wmma_f32_16x16x4_f32(false, A1, false, B0, (short)0, a0b, false, false);
            a1a = __builtin_amdgcn_wmma_f32_16x16x4_f32(false, A0, false, B1, (short)0, a1a, false, false);
            a1b = __builtin_amdgcn_wmma_f32_16x16x4_f32(false, A1, false, B1, (short)0, a1b, false, false);
            a2a = __builtin_amdgcn_wmma_f32_16x16x4_f32(false, A0, false, B2, (short)0, a2a, false, false);
            a2b = __builtin_amdgcn_wmma_f32_16x16x4_f32(false, A1, false, B2, (short)0, a2b, false, false);
            a3a = __builtin_amdgcn_wmma_f32_16x16x4_f32(false, A0, false, B3, (short)0, a3a, false, false);
            a3b = __builtin_amdgcn_wmma_f32_16x16x4_f32(false, A1, false, B3, (short)0, a3b, false, false);
        }
    }

    // C/D layout: VGPR v holds row M = v + 8*hi, column = l15 (+16 per d-tile)
#pragma unroll
    for (int v = 0; v < 8; ++v) {
        const int mrow = tile0 * 16 + v + 8 * hi;
        float* orow = out + (size_t)mrow * 64 + l15;
        float o0 = a0a[v], o1 = a1a[v], o2 = a2a[v], o3 = a3a[v];
        if (FUSE_RESIDUAL) {
            const float* rrow = resid + (size_t)mrow * 64 + l15;
            o0 += rrow[0]; o1 += rrow[16]; o2 += rrow[32]; o3 += rrow[48];
        }
        orow[0]  = fmaxf(o0, 0.0f);
        orow[16] = fmaxf(o1, 0.0f);
        orow[32] = fmaxf(o2, 0.0f);
        orow[48] = fmaxf(o3, 0.0f);
    }
    if (has1) {
#pragma unroll
        for (int v = 0; v < 8; ++v) {
            const int mrow = (tile0 + 1) * 16 + v + 8 * hi;
            float* orow = out + (size_t)mrow * 64 + l15;
            float o0 = a0b[v], o1 = a1b[v], o2 = a2b[v], o3 = a3b[v];
            if (FUSE_RESIDUAL) {
                const float* rrow = resid + (size_t)mrow * 64 + l15;
                o0 += rrow[0]; o1 += rrow[16]; o2 += rrow[32]; o3 += rrow[48];
            }
            orow[0]  = fmaxf(o0, 0.0f);
            orow[16] = fmaxf(o1, 0.0f);
            orow[32] = fmaxf(o2, 0.0f);
            orow[48] = fmaxf(o3, 0.0f);
        }
    }
}

extern "C" void kernel_launch(void* const* d_in, const int* in_sizes, int n_in,
                              void* d_out, int out_size, void* d_ws, size_t ws_size,
                              hipStream_t stream)
{
    const float* x    = (const float*)d_in[0];  // [N,64] fp32
    const int*   idx  = (const int*)  d_in[1];  // [N,K] int32
    const int*   mask = (const int*)  d_in[2];  // [N,K] bool->int
    const float* W1   = (const float*)d_in[3];  // [K,64,64]
    const float* W2   = (const float*)d_in[4];  // [K,64,64]
    float*       out  = (float*)d_out;          // [N,64]

    const int N = in_sizes[0] / 64;
    const int K = in_sizes[1] / N;

    // workspace: h [N*64 f32] | Wp1 [K*4096] | Wp2 [K*4096] | zrow [64]
    float* h    = (float*)d_ws;
    float* Wp1  = h + (size_t)N * 64;
    float* Wp2  = Wp1 + (size_t)K * 4096;
    float* zrow = Wp2 + (size_t)K * 4096;

    hipMemsetAsync(zrow, 0, 64 * sizeof(float), stream);  // zero row for masked gathers

    const int wtotal = K * 4096;
    repack_W<<<(wtotal + 255) / 256, 256, 0, stream>>>(W1, Wp1, wtotal);
    repack_W<<<(wtotal + 255) / 256, 256, 0, stream>>>(W2, Wp2, wtotal);

    const int numTiles = (N + 15) / 16;
    const int pairs    = (numTiles + 1) / 2;
    dim3 block(256);
    dim3 grid((pairs + WAVES_PER_BLOCK - 1) / WAVES_PER_BLOCK);

    // h = relu(conv1(x))
    sparse_conv_wmma<false><<<grid, block, 0, stream>>>(x, idx, mask, Wp1, zrow, nullptr, h, N, K);
    // out = relu(conv2(h) + x)
    sparse_conv_wmma<true ><<<grid, block, 0, stream>>>(h, idx, mask, Wp2, zrow, x, out, N, K);
}